// InformationTheoreticTopology_90091234000988
// MI455X (gfx1250) — compile-verified
//
#include <hip/hip_runtime.h>
#include <hip/hip_bf16.h>

// InformationTheoreticTopology loss for MI455X (gfx1250).
// Memory-bound streaming histogram + tiny MI finalize.
//   ws layout: uint32 hist[1024] | uint32 count | float esum   (4104 bytes)
// NOTE: itt_hist_kernel is first in the TU so the disasm snippet shows the
// async global->LDS pipeline (global_load_async_to_lds_b128 / s_wait_asynccnt).

#define NBINS 32
#define NB2   (NBINS * NBINS)
#define EPSF  1e-8f
#define TPB   256

// ---- CDNA5 async global->LDS path (guarded; falls back to direct loads) ----
#if defined(__HIP_DEVICE_COMPILE__) && __has_builtin(__builtin_amdgcn_global_load_async_to_lds_b128)
#define USE_ASYNC_LDS 1
// Builtin signature (probe-confirmed via round-1 diagnostic):
//   param0: v4i addrspace(1)* (global src), param1: v4i addrspace(3)* (LDS dst),
//   param2: imm offset, param3: imm cpol
typedef int v4i_t __attribute__((ext_vector_type(4)));
typedef __attribute__((address_space(1))) v4i_t as1_v4i;
typedef __attribute__((address_space(3))) v4i_t as3_v4i;
#else
#define USE_ASYNC_LDS 0
#endif

#if defined(__HIP_DEVICE_COMPILE__) && __has_builtin(__builtin_amdgcn_s_wait_asynccnt)
#define WAIT_ASYNCCNT(n) __builtin_amdgcn_s_wait_asynccnt(n)
#elif defined(__HIP_DEVICE_COMPILE__)
#define WAIT_ASYNCCNT(n) asm volatile("s_wait_asynccnt %0" ::"i"(n) : "memory")
#else
#define WAIT_ASYNCCNT(n)
#endif

__global__ __launch_bounds__(TPB) void itt_hist_kernel(
    const float* __restrict__ S, const float* __restrict__ T,
    const float* __restrict__ U, unsigned int* __restrict__ g_hist,
    unsigned int* __restrict__ g_cnt, float* __restrict__ g_esum,
    long long n, long long n4) {
  __shared__ unsigned int lhist[NB2];
  __shared__ float ls_esum;
  __shared__ unsigned int ls_cnt;
#if USE_ASYNC_LDS
  __shared__ float4 stage[3][2][TPB];  // [tensor][buffer][lane] : 24 KB
#endif

  const int tid = threadIdx.x;
  for (int i = tid; i < NB2; i += TPB) lhist[i] = 0u;
  if (tid == 0) { ls_esum = 0.0f; ls_cnt = 0u; }
  __syncthreads();

  float esum = 0.0f;
  unsigned int ecnt = 0u;

  auto procElem = [&](float s, float t, float u) {
    float topo = 1.0f - fabsf(s - t);
    float conf = 1.0f - u;
    int ti = (int)(topo * (float)NBINS);
    int ci = (int)(conf * (float)NBINS);
    ti = ti < 0 ? 0 : (ti > NBINS - 1 ? NBINS - 1 : ti);
    ci = ci < 0 ? 0 : (ci > NBINS - 1 ? NBINS - 1 : ci);
    bool valid = (topo >= 0.0f) && (topo < 1.0f) && (conf >= 0.0f) && (conf < 1.0f);
    if (valid) atomicAdd(&lhist[ti * NBINS + ci], 1u);
    if (conf > 0.8f) {
      float cl = fminf(fmaxf(topo, EPSF), 1.0f - EPSF);
      esum -= cl * logf(cl + EPSF);
      ecnt++;
    }
  };
  auto procVec = [&](float4 sv, float4 tv, float4 uv) {
    procElem(sv.x, tv.x, uv.x);
    procElem(sv.y, tv.y, uv.y);
    procElem(sv.z, tv.z, uv.z);
    procElem(sv.w, tv.w, uv.w);
  };

  const float4* S4 = (const float4*)S;
  const float4* T4 = (const float4*)T;
  const float4* U4 = (const float4*)U;

  const long long nChunks = n4 / TPB;  // full 256-float4 chunks
  const long long stride = (long long)gridDim.x;
  long long c = (long long)blockIdx.x;

#if USE_ASYNC_LDS
  // Double-buffered async DMA pipeline: global -> LDS, overlap DMA w/ binning.
  auto issue = [&](long long chunk, int buf) {
    long long base = chunk * TPB + tid;
    __builtin_amdgcn_global_load_async_to_lds_b128(
        (as1_v4i*)(S4 + base), (as3_v4i*)&stage[0][buf][tid], 0, 0);
    __builtin_amdgcn_global_load_async_to_lds_b128(
        (as1_v4i*)(T4 + base), (as3_v4i*)&stage[1][buf][tid], 0, 0);
    __builtin_amdgcn_global_load_async_to_lds_b128(
        (as1_v4i*)(U4 + base), (as3_v4i*)&stage[2][buf][tid], 0, 0);
  };
  if (c < nChunks) {
    int buf = 0;
    issue(c, buf);
    for (;;) {
      long long cn = c + stride;
      if (cn < nChunks) {
        issue(cn, buf ^ 1);   // 6 outstanding; wait for first 3 (in-order)
        WAIT_ASYNCCNT(3);
      } else {
        WAIT_ASYNCCNT(0);
      }
      float4 sv = stage[0][buf][tid];
      float4 tv = stage[1][buf][tid];
      float4 uv = stage[2][buf][tid];
      procVec(sv, tv, uv);
      if (cn >= nChunks) break;
      c = cn;
      buf ^= 1;
    }
  }
#else
  for (; c < nChunks; c += stride) {
    long long base = c * TPB + tid;
    procVec(S4[base], T4[base], U4[base]);
  }
#endif

  // Tails (empty for the 4096x8192 reference sizes, kept for generality).
  if (blockIdx.x == 0) {
    long long i = nChunks * TPB + tid;
    if (i < n4) procVec(S4[i], T4[i], U4[i]);
    if (tid == 0) {
      for (long long j = n4 * 4; j < n; ++j) procElem(S[j], T[j], U[j]);
    }
  }

  // Block-level reduction of entropy accumulators.
  if (ecnt) {
    atomicAdd(&ls_esum, esum);
    atomicAdd(&ls_cnt, ecnt);
  }
  __syncthreads();

  // Merge privatized histogram into global workspace.
  for (int i = tid; i < NB2; i += TPB) {
    unsigned int v = lhist[i];
    if (v) atomicAdd(&g_hist[i], v);
  }
  if (tid == 0 && ls_cnt) {
    atomicAdd(g_cnt, ls_cnt);
    atomicAdd(g_esum, ls_esum);
  }
}

__global__ void itt_init_kernel(unsigned int* __restrict__ g_hist,
                                unsigned int* __restrict__ g_cnt,
                                float* __restrict__ g_esum) {
  int i = blockIdx.x * blockDim.x + threadIdx.x;
  if (i < NB2) g_hist[i] = 0u;
  if (i == 0) { *g_cnt = 0u; *g_esum = 0.0f; }
}

__global__ __launch_bounds__(NB2) void itt_finalize_kernel(
    const unsigned int* __restrict__ g_hist, const unsigned int* __restrict__ g_cnt,
    const float* __restrict__ g_esum, float* __restrict__ out) {
  __shared__ float jp[NB2];
  __shared__ float red[NB2];
  __shared__ float mt[NBINS];
  __shared__ float mc[NBINS];
  __shared__ float s_tot;

  const int tid = threadIdx.x;
  float h = (float)g_hist[tid];
  red[tid] = h;
  __syncthreads();
  for (int s = NB2 / 2; s > 0; s >>= 1) {
    if (tid < s) red[tid] += red[tid + s];
    __syncthreads();
  }
  if (tid == 0) s_tot = red[0];
  __syncthreads();

  const float total = s_tot;
  const float p = h / (total + EPSF);
  jp[tid] = p;
  __syncthreads();

  if (tid < NBINS) {  // marginal over topology rows
    float s = 0.0f;
    for (int cc = 0; cc < NBINS; ++cc) s += jp[tid * NBINS + cc];
    mt[tid] = s;
  } else if (tid < 2 * NBINS) {  // marginal over confidence cols
    int cc = tid - NBINS;
    float s = 0.0f;
    for (int r = 0; r < NBINS; ++r) s += jp[r * NBINS + cc];
    mc[cc] = s;
  }
  __syncthreads();

  const int r = tid >> 5;
  const int cidx = tid & (NBINS - 1);
  float ratio = p / (mt[r] * mc[cidx] + EPSF) + EPSF;
  float term = (p > EPSF) ? p * logf(ratio) : 0.0f;
  red[tid] = term;
  __syncthreads();
  for (int s = NB2 / 2; s > 0; s >>= 1) {
    if (tid < s) red[tid] += red[tid + s];
    __syncthreads();
  }

  if (tid == 0) {
    float mi = (total < EPSF) ? 0.0f : red[0];
    float mi_loss = -mi;
    unsigned int cnt = *g_cnt;
    float es = *g_esum;
    float denom = (float)(cnt > 1u ? cnt : 1u);
    float entropy_loss = (cnt > 10u) ? (es / denom) : 0.0f;
    out[0] = mi_loss + 0.1f * entropy_loss;
  }
}

extern "C" void kernel_launch(void* const* d_in, const int* in_sizes, int n_in,
                              void* d_out, int out_size, void* d_ws, size_t ws_size,
                              hipStream_t stream) {
  (void)n_in; (void)out_size; (void)ws_size;
  const float* S = (const float*)d_in[0];
  const float* T = (const float*)d_in[1];
  const float* U = (const float*)d_in[2];
  const long long n = (long long)in_sizes[0];
  const long long n4 = n / 4;

  unsigned int* g_hist = (unsigned int*)d_ws;        // NB2 uints
  unsigned int* g_cnt  = g_hist + NB2;               // 1 uint
  float*        g_esum = (float*)(g_hist + NB2 + 1); // 1 float

  itt_init_kernel<<<(NB2 + TPB - 1) / TPB, TPB, 0, stream>>>(g_hist, g_cnt, g_esum);

  long long nChunks = n4 / TPB;
  int blocks = 2048;  // ~8 WGs/WGP on a ~256-WGP part; deep enough pipeline
  if (nChunks < (long long)blocks) blocks = (int)(nChunks > 0 ? nChunks : 1);
  itt_hist_kernel<<<blocks, TPB, 0, stream>>>(S, T, U, g_hist, g_cnt, g_esum, n, n4);

  itt_finalize_kernel<<<1, NB2, 0, stream>>>(g_hist, g_cnt, g_esum, (float*)d_out);
}